// LinearAttention_48206712930860
// MI455X (gfx1250) — compile-verified
//
#include <hip/hip_runtime.h>
#include <math.h>

// Causal linear attention (elu+1 feature map) on CDNA5 WMMA, chunked + segmented.
// Shapes fixed to the reference: B=4, L=2048, H=16, D=M=64, fp32.
//
// State is additive across the sequence, so split L into NS segments:
//  phase1: per-segment KV = Kf^T V and ksum column sums   (WMMA)
//  phase2: in-place exclusive prefix over segments        (tiny)
//  phase3: chunked scan per segment seeded with prefix    (WMMA)
// Falls back to the single-pass scan if the workspace is too small.

typedef __attribute__((ext_vector_type(2))) float v2f;
typedef __attribute__((ext_vector_type(8))) float v8f;

#define BB   4
#define LL   2048
#define HH   16
#define DD   64
#define CCH  16
#define NS   8                 // segments per (b,h)
#define SEGL (LL / NS)         // 256 rows per segment
#define CHPS (SEGL / CCH)      // 16 chunks per segment
#define NCH  (LL / CCH)        // 128 chunks (one-pass fallback)
#define FEPS 1e-6f

__device__ __forceinline__ float fmap(float x) {
    return x > 0.f ? x + 1.f : __expf(x);   // elu(x)+1
}

__device__ __forceinline__ v8f wmma4(v2f a, v2f b, v8f c) {
    return __builtin_amdgcn_wmma_f32_16x16x4_f32(
        false, a, false, b, (short)0, c, false, false);
}

// ---------------- LDS float offsets (phase3 / one-pass) ----------------
#define OFF_QF   0        // Qf 16x64 row-major
#define OFF_KF   1024     // Kf 16x64 row-major
#define OFF_KPI  2048     // Kf pair-interleaved over rows
#define OFF_VPI  3072     // V  pair-interleaved over rows
#define OFF_S    4096     // 4 waves x (64x16 pair-interleaved) state mirror
#define OFF_A    8192     // masked Qf@Kf^T 16x16 row-major
#define OFF_KS   8448     // running column-sum of Kf (64)
#define OFF_KD   8512     // this-chunk column-sum of Kf (64)
#define OFF_Z    8576     // normalizer (16)
#define SMEM_F   8592

// =====================================================================
// Phase 1: per-(b,h,seg) KV = Kf^T @ V (64x64) and ksum (64)
// =====================================================================
__global__ __launch_bounds__(128)
void linattn_phase1(const float* __restrict__ k,
                    const float* __restrict__ v,
                    float* __restrict__ segKV,
                    float* __restrict__ segKS) {
    __shared__ float sm[3072];
    const int tid  = threadIdx.x;
    const int lane = tid & 31;
    const int wave = tid >> 5;
    const int lh   = lane >> 4;
    const int lc   = lane & 15;
    const int bh   = blockIdx.x / NS;
    const int seg  = blockIdx.x % NS;
    const int b    = bh / HH;
    const int h    = bh % HH;

    float* s_kf  = sm;          // 16x64 row-major
    float* s_kpi = sm + 1024;   // pair-interleaved
    float* s_vpi = sm + 2048;   // pair-interleaved

    const v8f zero8 = {0.f,0.f,0.f,0.f,0.f,0.f,0.f,0.f};
    v8f st[4];
    #pragma unroll
    for (int t = 0; t < 4; ++t) st[t] = zero8;
    float ks0 = 0.f, ks1 = 0.f;           // wave2 column-sum accumulators

    const int row = tid >> 3;
    const int c0  = (tid & 7) * 8;

    for (int ch = 0; ch < CHPS; ++ch) {
        const int l0 = seg * SEGL + ch * CCH;
        __syncthreads();
        {
            size_t g = ((size_t)((b * LL + l0 + row) * HH + h)) * DD + c0;
            float ka8[8], va8[8];
            *(float4*)(ka8)     = *(const float4*)(k + g);
            *(float4*)(ka8 + 4) = *(const float4*)(k + g + 4);
            *(float4*)(va8)     = *(const float4*)(v + g);
            *(float4*)(va8 + 4) = *(const float4*)(v + g + 4);
            #pragma unroll
            for (int j = 0; j < 8; ++j) {
                int c  = c0 + j;
                float kf = fmap(ka8[j]);
                int pi = ((row >> 1) * 64 + c) * 2 + (row & 1);
                s_kf[row * 64 + c] = kf;
                s_kpi[pi] = kf;
                s_vpi[pi] = va8[j];
            }
            if (ch + 1 < CHPS) {
                size_t gn = g + (size_t)CCH * HH * DD;
                __builtin_prefetch(k + gn, 0, 3);
                __builtin_prefetch(v + gn, 0, 3);
            }
        }
        __syncthreads();

        v2f bv[4];
        #pragma unroll
        for (int kk = 0; kk < 4; ++kk) {
            int p = 2 * kk + lh;
            bv[kk] = *(const v2f*)(s_vpi + (p * 64 + 16 * wave + lc) * 2);
        }
        #pragma unroll
        for (int t = 0; t < 4; ++t) {
            #pragma unroll
            for (int kk = 0; kk < 4; ++kk) {
                int p = 2 * kk + lh;
                v2f ak = *(const v2f*)(s_kpi + (p * 64 + 16 * t + lc) * 2);
                st[t] = wmma4(ak, bv[kk], st[t]);
            }
        }
        if (wave == 2) {
            float a0 = 0.f, a1 = 0.f;
            #pragma unroll
            for (int r16 = 0; r16 < 16; ++r16) {
                a0 += s_kf[r16 * 64 + lane * 2];
                a1 += s_kf[r16 * 64 + lane * 2 + 1];
            }
            ks0 += a0; ks1 += a1;
        }
    }

    // store KV (row-major 64x64) and ksum for this segment
    float* kvout = segKV + (size_t)blockIdx.x * 4096;
    #pragma unroll
    for (int t = 0; t < 4; ++t) {
        #pragma unroll
        for (int r = 0; r < 8; ++r) {
            int srow = 16 * t + r + 8 * lh;
            kvout[srow * 64 + 16 * wave + lc] = st[t][r];
        }
    }
    if (wave == 2) {
        segKS[(size_t)blockIdx.x * 64 + lane * 2]     = ks0;
        segKS[(size_t)blockIdx.x * 64 + lane * 2 + 1] = ks1;
    }
}

// =====================================================================
// Phase 2: in-place exclusive prefix over the NS segments of each (b,h)
// =====================================================================
__global__ __launch_bounds__(128)
void linattn_phase2(float* __restrict__ segKV,
                    float* __restrict__ segKS) {
    const int bh  = blockIdx.x;
    const int tid = threadIdx.x;
    for (int e = tid; e < 4096; e += 128) {
        float acc = 0.f;
        #pragma unroll
        for (int s = 0; s < NS; ++s) {
            size_t idx = ((size_t)(bh * NS + s)) * 4096 + e;
            float t = segKV[idx];
            segKV[idx] = acc;
            acc += t;
        }
    }
    if (tid < 64) {
        float acc = 0.f;
        #pragma unroll
        for (int s = 0; s < NS; ++s) {
            size_t idx = ((size_t)(bh * NS + s)) * 64 + tid;
            float t = segKS[idx];
            segKS[idx] = acc;
            acc += t;
        }
    }
}

// =====================================================================
// Phase 3 / one-pass: chunked causal scan. seeded==true loads the
// segment-initial state from the workspace; otherwise starts from zero
// and scans the whole sequence (fallback).
// =====================================================================
template <bool SEEDED>
__global__ __launch_bounds__(128)
void linattn_scan(const float* __restrict__ q,
                  const float* __restrict__ k,
                  const float* __restrict__ v,
                  float* __restrict__ out,
                  const float* __restrict__ segKV,
                  const float* __restrict__ segKS) {
    __shared__ float sm[SMEM_F];
    const int tid  = threadIdx.x;
    const int lane = tid & 31;
    const int wave = tid >> 5;
    const int lh   = lane >> 4;
    const int lc   = lane & 15;
    const int bh   = SEEDED ? (blockIdx.x / NS) : blockIdx.x;
    const int seg  = SEEDED ? (blockIdx.x % NS) : 0;
    const int b    = bh / HH;
    const int h    = bh % HH;
    const int nch  = SEEDED ? CHPS : NCH;

    float* s_qf  = sm + OFF_QF;
    float* s_kf  = sm + OFF_KF;
    float* s_kpi = sm + OFF_KPI;
    float* s_vpi = sm + OFF_VPI;
    float* s_spi = sm + OFF_S + wave * 1024;
    float* s_a   = sm + OFF_A;
    float* s_ks  = sm + OFF_KS;
    float* s_kd  = sm + OFF_KD;
    float* s_z   = sm + OFF_Z;

    const v8f zero8 = {0.f,0.f,0.f,0.f,0.f,0.f,0.f,0.f};
    v8f st[4];

    if (SEEDED) {
        const float* kvin = segKV + (size_t)blockIdx.x * 4096;
        #pragma unroll
        for (int t = 0; t < 4; ++t) {
            #pragma unroll
            for (int r = 0; r < 8; ++r) {
                int srow = 16 * t + r + 8 * lh;
                float val = kvin[srow * 64 + 16 * wave + lc];
                st[t][r] = val;
                s_spi[((srow >> 1) * 16 + lc) * 2 + (srow & 1)] = val;
            }
        }
        if (tid < 64) s_ks[tid] = segKS[(size_t)blockIdx.x * 64 + tid];
    } else {
        #pragma unroll
        for (int t = 0; t < 4; ++t) st[t] = zero8;
        #pragma unroll
        for (int i = 0; i < 32; ++i) sm[OFF_S + tid + i * 128] = 0.f;
        if (tid < 64) s_ks[tid] = 0.f;
    }

    const int row = tid >> 3;
    const int c0  = (tid & 7) * 8;

    for (int ch = 0; ch < nch; ++ch) {
        const int l0 = seg * SEGL + ch * CCH;
        __syncthreads();

        // ---- cooperative load: global -> LDS with feature map ----
        {
            size_t g = ((size_t)((b * LL + l0 + row) * HH + h)) * DD + c0;
            float qa8[8], ka8[8], va8[8];
            *(float4*)(qa8)     = *(const float4*)(q + g);
            *(float4*)(qa8 + 4) = *(const float4*)(q + g + 4);
            *(float4*)(ka8)     = *(const float4*)(k + g);
            *(float4*)(ka8 + 4) = *(const float4*)(k + g + 4);
            *(float4*)(va8)     = *(const float4*)(v + g);
            *(float4*)(va8 + 4) = *(const float4*)(v + g + 4);
            #pragma unroll
            for (int j = 0; j < 8; ++j) {
                int c  = c0 + j;
                float qf = fmap(qa8[j]);
                float kf = fmap(ka8[j]);
                int pi = ((row >> 1) * 64 + c) * 2 + (row & 1);
                s_qf[row * 64 + c] = qf;
                s_kf[row * 64 + c] = kf;
                s_kpi[pi] = kf;
                s_vpi[pi] = va8[j];
            }
            if (ch + 1 < nch) {
                size_t gn = g + (size_t)CCH * HH * DD;
                __builtin_prefetch(q + gn, 0, 3);
                __builtin_prefetch(k + gn, 0, 3);
                __builtin_prefetch(v + gn, 0, 3);
            }
        }
        __syncthreads();

        // ---- Qf A-operands ----
        v2f qa[16];
        #pragma unroll
        for (int s = 0; s < 16; ++s)
            qa[s] = *(const v2f*)(s_qf + lc * 64 + 4 * s + 2 * lh);

        // ---- inter-chunk: oc = Qf @ S_prev ----
        v8f oc = zero8;
        #pragma unroll
        for (int t = 0; t < 4; ++t) {
            #pragma unroll
            for (int kk = 0; kk < 4; ++kk) {
                int p = 8 * t + 2 * kk + lh;
                v2f bs = *(const v2f*)(s_spi + (p * 16 + lc) * 2);
                oc = wmma4(qa[4 * t + kk], bs, oc);
            }
        }

        // ---- wave0: A = Qf @ Kf^T, causal mask, publish ----
        if (wave == 0) {
            v8f aq = zero8;
            #pragma unroll
            for (int s = 0; s < 16; ++s) {
                v2f bk = *(const v2f*)(s_kf + lc * 64 + 4 * s + 2 * lh);
                aq = wmma4(qa[s], bk, aq);
            }
            #pragma unroll
            for (int r = 0; r < 8; ++r) {
                int m = r + 8 * lh;
                s_a[m * 16 + lc] = (lc <= m) ? aq[r] : 0.f;
            }
        }
        __syncthreads();

        // ---- wave1: normalizer z ; wave2: chunk column-sum of Kf ----
        if (wave == 1 && lane < 16) {
            float zz = FEPS;
            #pragma unroll
            for (int n = 0; n < 16; ++n) zz += s_a[lane * 16 + n];
            #pragma unroll 8
            for (int d = 0; d < 64; ++d) zz += s_qf[lane * 64 + d] * s_ks[d];
            s_z[lane] = zz;
        }
        if (wave == 2) {
            #pragma unroll
            for (int j = 0; j < 2; ++j) {
                int d = lane * 2 + j;
                float ssum = 0.f;
                #pragma unroll
                for (int r16 = 0; r16 < 16; ++r16) ssum += s_kf[r16 * 64 + d];
                s_kd[d] = ssum;
            }
        }

        // ---- V B-operands ----
        v2f bv[4];
        #pragma unroll
        for (int kk = 0; kk < 4; ++kk) {
            int p = 2 * kk + lh;
            bv[kk] = *(const v2f*)(s_vpi + (p * 64 + 16 * wave + lc) * 2);
        }
        // ---- intra-chunk: oc += tril(A) @ V ----
        #pragma unroll
        for (int kk = 0; kk < 4; ++kk) {
            v2f av = *(const v2f*)(s_a + lc * 16 + 4 * kk + 2 * lh);
            oc = wmma4(av, bv[kk], oc);
        }
        // ---- state update: S += Kf^T @ V ----
        #pragma unroll
        for (int t = 0; t < 4; ++t) {
            #pragma unroll
            for (int kk = 0; kk < 4; ++kk) {
                int p = 2 * kk + lh;
                v2f ak = *(const v2f*)(s_kpi + (p * 64 + 16 * t + lc) * 2);
                st[t] = wmma4(ak, bv[kk], st[t]);
            }
        }
        __syncthreads();

        // ---- scale by 1/z and store output slab ----
        #pragma unroll
        for (int r = 0; r < 8; ++r) {
            int m = r + 8 * lh;
            float zz = s_z[m];
            size_t go = ((size_t)((b * LL + l0 + m) * HH + h)) * DD
                        + 16 * wave + lc;
            out[go] = oc[r] / zz;
        }
        // ---- mirror updated S to LDS for next chunk ----
        #pragma unroll
        for (int t = 0; t < 4; ++t) {
            #pragma unroll
            for (int r = 0; r < 8; ++r) {
                int srow = 16 * t + r + 8 * lh;
                s_spi[((srow >> 1) * 16 + lc) * 2 + (srow & 1)] = st[t][r];
            }
        }
        // ---- wave3: Ksum += chunk column-sum ----
        if (wave == 3) {
            s_ks[lane * 2]     += s_kd[lane * 2];
            s_ks[lane * 2 + 1] += s_kd[lane * 2 + 1];
        }
    }
}

extern "C" void kernel_launch(void* const* d_in, const int* in_sizes, int n_in,
                              void* d_out, int out_size, void* d_ws, size_t ws_size,
                              hipStream_t stream) {
    (void)in_sizes; (void)n_in; (void)out_size;
    const float* q = (const float*)d_in[0];
    const float* k = (const float*)d_in[1];
    const float* v = (const float*)d_in[2];
    float* out = (float*)d_out;

    const size_t nSeg   = (size_t)BB * HH * NS;
    const size_t needWs = nSeg * (4096 + 64) * sizeof(float);

    if (ws_size >= needWs) {
        float* segKV = (float*)d_ws;
        float* segKS = segKV + nSeg * 4096;
        hipLaunchKernelGGL(linattn_phase1, dim3(BB * HH * NS), dim3(128), 0,
                           stream, k, v, segKV, segKS);
        hipLaunchKernelGGL(linattn_phase2, dim3(BB * HH), dim3(128), 0,
                           stream, segKV, segKS);
        hipLaunchKernelGGL((linattn_scan<true>), dim3(BB * HH * NS), dim3(128),
                           0, stream, q, k, v, out, segKV, segKS);
    } else {
        hipLaunchKernelGGL((linattn_scan<false>), dim3(BB * HH), dim3(128),
                           0, stream, q, k, v, out,
                           (const float*)nullptr, (const float*)nullptr);
    }
}